// CCAMDec_2903397892317
// MI455X (gfx1250) — compile-verified
//
#include <hip/hip_runtime.h>
#include <cmath>

typedef __attribute__((ext_vector_type(2))) float v2f;
typedef __attribute__((ext_vector_type(8))) float v8f;

#define N_B   8
#define C_DIM 512
#define K_DIM 64
#define HW_DIM 4096
#define CT    (C_DIM / 16)        // 32 c-tiles of 16 rows
#define WAVES_A 16
#define THREADS_A (WAVES_A * 32)
#define S_PER_WAVE (HW_DIM / WAVES_A)   // 256
#define WAVES_B 8
#define SG      (HW_DIM / (64 * WAVES_B)) // 8 s-groups

// ---------------------------------------------------------------------------
// Kernel A: energy E = X·Yᵀ (per 16-row c-tile, full K=64) via f32 WMMA,
// cross-wave LDS reduction, inverted softmax, write attention to ws.
// 16 waves/block, unroll-4 s-loop -> deep VMEM pipeline to saturate HBM.
// ---------------------------------------------------------------------------
__global__ __launch_bounds__(THREADS_A)
void cca_energy_softmax(const float* __restrict__ x,
                        const float* __restrict__ y,
                        float* __restrict__ attn) {
    __shared__ float part[WAVES_A][4][8][32];   // 64 KB partial E
    __shared__ float Ered[16][64];              // 4 KB reduced E

    const int blk   = blockIdx.x;
    const int n     = blk / CT;
    const int c0    = (blk % CT) * 16;
    const int tid   = threadIdx.x;
    const int w     = tid >> 5;
    const int lane  = tid & 31;
    const int laneM = lane & 15;
    const int hi    = lane >> 4;               // 0: K=0,1  1: K=2,3

    const float* xrow  = x + (size_t)(n * C_DIM + c0 + laneM) * HW_DIM;
    const float* ybase = y + (size_t)(n * K_DIM) * HW_DIM;

    v8f acc0 = {}, acc1 = {}, acc2 = {}, acc3 = {};

    const int sBeg = w * S_PER_WAVE;
    const int sEnd = sBeg + S_PER_WAVE;
    #pragma unroll 4
    for (int s = sBeg; s < sEnd; s += 4) {
        const int so = s + 2 * hi;
        // A frag: X[c0+laneM][so..so+1]  (8B-aligned b64 load)
        v2f a  = *(const v2f*)(xrow + so);
        // B frags: B[s_local][k] = Y[kt*16+k][s+s_local]
        v2f b0 = *(const v2f*)(ybase + (size_t)( 0 + laneM) * HW_DIM + so);
        v2f b1 = *(const v2f*)(ybase + (size_t)(16 + laneM) * HW_DIM + so);
        v2f b2 = *(const v2f*)(ybase + (size_t)(32 + laneM) * HW_DIM + so);
        v2f b3 = *(const v2f*)(ybase + (size_t)(48 + laneM) * HW_DIM + so);
        acc0 = __builtin_amdgcn_wmma_f32_16x16x4_f32(false, a, false, b0, (short)0, acc0, false, false);
        acc1 = __builtin_amdgcn_wmma_f32_16x16x4_f32(false, a, false, b1, (short)0, acc1, false, false);
        acc2 = __builtin_amdgcn_wmma_f32_16x16x4_f32(false, a, false, b2, (short)0, acc2, false, false);
        acc3 = __builtin_amdgcn_wmma_f32_16x16x4_f32(false, a, false, b3, (short)0, acc3, false, false);
    }

    #pragma unroll
    for (int j = 0; j < 8; ++j) {
        part[w][0][j][lane] = acc0[j];
        part[w][1][j][lane] = acc1[j];
        part[w][2][j][lane] = acc2[j];
        part[w][3][j][lane] = acc3[j];
    }
    __syncthreads();

    // Reduce wave-partials into Ered[M][k] (C/D layout: M=j+8*hi, N=laneM)
    for (int idx = tid; idx < 16 * 64; idx += THREADS_A) {
        const int M  = idx >> 6;
        const int k  = idx & 63;
        const int kt = k >> 4;
        const int j  = M & 7;
        const int l  = (k & 15) + ((M >> 3) << 4);
        float s = 0.f;
        #pragma unroll
        for (int ww = 0; ww < WAVES_A; ++ww) s += part[ww][kt][j][l];
        Ered[M][k] = s;
    }
    __syncthreads();

    // Inverted softmax: softmax(max-E) == exp(minE - E)/sum. One row per thread.
    if (tid < 16) {
        float mn = Ered[tid][0];
        #pragma unroll
        for (int k = 1; k < 64; ++k) mn = fminf(mn, Ered[tid][k]);
        float sum = 0.f;
        #pragma unroll
        for (int k = 0; k < 64; ++k) sum += __expf(mn - Ered[tid][k]);
        const float inv = 1.0f / sum;
        float* arow = attn + (size_t)(n * C_DIM + c0 + tid) * K_DIM;
        #pragma unroll
        for (int k = 0; k < 64; ++k) arow[k] = __expf(mn - Ered[tid][k]) * inv;
    }
}

// ---------------------------------------------------------------------------
// Kernel B: O = P·Y via f32 WMMA, fused residual out = x + scale*O.
// A-fragments (attention row strip) loaded once per wave, reused for 4 s-tiles.
// ---------------------------------------------------------------------------
__global__ __launch_bounds__(256)
void cca_out(const float* __restrict__ x, const float* __restrict__ y,
             const float* __restrict__ attn, const float* __restrict__ scale_p,
             float* __restrict__ out) {
    const int blk   = blockIdx.x;         // = n*CT*SG + ct*SG + sg
    const int sg    = blk & (SG - 1);
    const int ct    = (blk >> 3) & (CT - 1);
    const int n     = blk >> 8;
    const int tid   = threadIdx.x;
    const int w     = tid >> 5;
    const int lane  = tid & 31;
    const int laneM = lane & 15;
    const int hi    = lane >> 4;
    const int c0    = ct * 16;
    const int s0    = sg * (64 * WAVES_B) + w * 64;

    const float scale = scale_p[0];

    // A fragments: 16 k-steps of 4. lane<16: k=4t,4t+1 ; lane>=16: k=4t+2,4t+3
    const float* arow = attn + (size_t)(n * C_DIM + c0 + laneM) * K_DIM + 2 * hi;
    v2f af[16];
    #pragma unroll
    for (int st = 0; st < 16; ++st) af[st] = *(const v2f*)(arow + 4 * st);

    const float* yb = y + (size_t)(n * K_DIM) * HW_DIM;

    #pragma unroll
    for (int t = 0; t < 4; ++t) {
        const int sc = s0 + t * 16 + laneM;
        v8f acc = {};
        #pragma unroll
        for (int st = 0; st < 16; ++st) {
            const int kr = 4 * st + 2 * hi;
            v2f b;
            b.x = yb[(size_t)kr * HW_DIM + sc];
            b.y = yb[(size_t)(kr + 1) * HW_DIM + sc];
            acc = __builtin_amdgcn_wmma_f32_16x16x4_f32(false, af[st], false, b, (short)0, acc, false, false);
        }
        #pragma unroll
        for (int j = 0; j < 8; ++j) {
            const int row = c0 + j + 8 * hi;
            const size_t idx = (size_t)(n * C_DIM + row) * HW_DIM + (size_t)(s0 + t * 16 + laneM);
            out[idx] = x[idx] + scale * acc[j];
        }
    }
}

extern "C" void kernel_launch(void* const* d_in, const int* in_sizes, int n_in,
                              void* d_out, int out_size, void* d_ws, size_t ws_size,
                              hipStream_t stream) {
    const float* x     = (const float*)d_in[0];
    const float* y     = (const float*)d_in[1];
    const float* scale = (const float*)d_in[2];
    float*       out   = (float*)d_out;
    float*       attn  = (float*)d_ws;   // N*C*K floats = 1 MB scratch

    cca_energy_softmax<<<N_B * CT, THREADS_A, 0, stream>>>(x, y, attn);
    cca_out<<<N_B * CT * SG, 256, 0, stream>>>(x, y, attn, scale, out);
}